// GraphPropagation_25357486915690
// MI455X (gfx1250) — compile-verified
//
#include <hip/hip_runtime.h>
#include <stdint.h>

#define D_FEAT 128
#define WPB 8        // wave32s per block in the propagation kernel (1 node per wave)
#define CHUNK 64     // edge ids staged into LDS per round per wave

__global__ __launch_bounds__(256) void zero_i32(int* __restrict__ p, int n) {
  int i = blockIdx.x * 256 + threadIdx.x;
  if (i < n) p[i] = 0;
}

__global__ __launch_bounds__(256) void build_deg(const int* __restrict__ dst,
                                                 int* __restrict__ deg, int e) {
  int i = blockIdx.x * 256 + threadIdx.x;
  if (i < e) atomicAdd(&deg[dst[i]], 1);
}

// Single-block exclusive scan of deg[0..n) -> rowptr[0..n], cursor copy.
__global__ __launch_bounds__(1024) void scan_deg(const int* __restrict__ deg,
                                                 int* __restrict__ rowptr,
                                                 int* __restrict__ cursor, int n) {
  __shared__ int part[1024];
  const int t = threadIdx.x;
  const int chunk = (n + 1023) >> 10;
  const int b = t * chunk;
  const int e = (b + chunk < n) ? (b + chunk) : n;
  int sum = 0;
  for (int i = b; i < e; ++i) sum += deg[i];
  part[t] = sum;
  __syncthreads();
  for (int off = 1; off < 1024; off <<= 1) {   // Hillis-Steele inclusive scan
    int v = (t >= off) ? part[t - off] : 0;
    __syncthreads();
    part[t] += v;
    __syncthreads();
  }
  int run = (t == 0) ? 0 : part[t - 1];
  for (int i = b; i < e; ++i) { rowptr[i] = run; cursor[i] = run; run += deg[i]; }
  if (t == 1023) rowptr[n] = part[1023];
}

__global__ __launch_bounds__(256) void fill_csr(const int* __restrict__ dst,
                                                int* __restrict__ cursor,
                                                int* __restrict__ csr, int e) {
  int i = blockIdx.x * 256 + threadIdx.x;
  if (i < e) { int slot = atomicAdd(&cursor[dst[i]], 1); csr[slot] = i; }
}

// Sort each node's edge-id list (unique keys -> deterministic order regardless
// of atomic races in fill_csr), then resolve edge id -> src node id.
__global__ __launch_bounds__(256) void sort_resolve(const int* __restrict__ rowptr,
                                                    int* __restrict__ csr,
                                                    const int* __restrict__ src, int n) {
  int i = blockIdx.x * 256 + threadIdx.x;
  if (i >= n) return;
  const int s0 = rowptr[i], s1 = rowptr[i + 1];
  for (int a = s0 + 1; a < s1; ++a) {
    int key = csr[a]; int b = a - 1;
    while (b >= s0 && csr[b] > key) { csr[b + 1] = csr[b]; --b; }
    csr[b + 1] = key;
  }
  for (int a = s0; a < s1; ++a) csr[a] = src[csr[a]];
}

// Fused iteration: hnext[i] = (1-a)*norm[i] * sum_{e: dst=i} hcur[src_e]*norm[src_e] + a*h0[i]
// One wave per node; lane owns float4 (128 = 32*4). The src-id list is staged
// into LDS with gfx1250 async global->LDS loads; the gather loop is unrolled
// by two edges with independent accumulators so two global_load_b128 row
// fetches are in flight per wave.
__global__ __launch_bounds__(WPB * 32) void prop(const float* __restrict__ hcur,
                                                 const float* __restrict__ h0,
                                                 const float* __restrict__ norm,
                                                 const int* __restrict__ rowptr,
                                                 const int* __restrict__ csr,
                                                 float* __restrict__ hnext,
                                                 int n, float alpha) {
  __shared__ int sid[WPB][CHUNK];
  const int wave = threadIdx.x >> 5;
  const int lane = threadIdx.x & 31;
  const int node = blockIdx.x * WPB + wave;
  if (node >= n) return;
  const int s0 = rowptr[node], s1 = rowptr[node + 1];
  const unsigned lds_base = (unsigned)(uintptr_t)(&sid[wave][0]);  // LDS byte offset
  const int col = lane * 4;
  float4 acc0 = make_float4(0.f, 0.f, 0.f, 0.f);
  float4 acc1 = make_float4(0.f, 0.f, 0.f, 0.f);

  for (int base = s0; base < s1; base += CHUNK) {
    const int cnt = (s1 - base < CHUNK) ? (s1 - base) : CHUNK;
    const int rounds = (cnt + 31) & ~31;
    // async-copy cnt src ids into this wave's LDS slice (32 dwords / instruction)
    for (int j = lane; j < rounds; j += 32) {
      const int idx = base + ((j < cnt) ? j : (cnt - 1));   // clamp tail lanes in-bounds
      const unsigned ldsa = lds_base + 4u * (unsigned)j;
      const unsigned long long ga = (unsigned long long)(uintptr_t)(csr + idx);
      asm volatile("global_load_async_to_lds_b32 %0, %1, off"
                   :: "v"(ldsa), "v"(ga) : "memory");
    }
    asm volatile("s_wait_asynccnt 0x0" ::: "memory");

    int j = 0;
    for (; j + 1 < cnt; j += 2) {            // two gathers in flight per wave
      const int sa = __builtin_amdgcn_readfirstlane(sid[wave][j]);
      const int sb = __builtin_amdgcn_readfirstlane(sid[wave][j + 1]);
      const float na = norm[sa];
      const float nb = norm[sb];
      const float4 va = *(const float4*)(hcur + (size_t)sa * D_FEAT + col);  // b128
      const float4 vb = *(const float4*)(hcur + (size_t)sb * D_FEAT + col);  // b128
      acc0.x += va.x * na; acc0.y += va.y * na; acc0.z += va.z * na; acc0.w += va.w * na;
      acc1.x += vb.x * nb; acc1.y += vb.y * nb; acc1.z += vb.z * nb; acc1.w += vb.w * nb;
    }
    if (j < cnt) {                           // odd tail
      const int sa = __builtin_amdgcn_readfirstlane(sid[wave][j]);
      const float na = norm[sa];
      const float4 va = *(const float4*)(hcur + (size_t)sa * D_FEAT + col);
      acc0.x += va.x * na; acc0.y += va.y * na; acc0.z += va.z * na; acc0.w += va.w * na;
    }
  }

  const float g = norm[node] * (1.0f - alpha);
  const float4 hz = *(const float4*)(h0 + (size_t)node * D_FEAT + col);
  float4 o;
  o.x = (acc0.x + acc1.x) * g + alpha * hz.x;
  o.y = (acc0.y + acc1.y) * g + alpha * hz.y;
  o.z = (acc0.z + acc1.z) * g + alpha * hz.z;
  o.w = (acc0.w + acc1.w) * g + alpha * hz.w;
  *(float4*)(hnext + (size_t)node * D_FEAT + col) = o;
}

extern "C" void kernel_launch(void* const* d_in, const int* in_sizes, int n_in,
                              void* d_out, int out_size, void* d_ws, size_t ws_size,
                              hipStream_t stream) {
  const float* h_in = (const float*)d_in[0];   // [N,128]
  const float* norm = (const float*)d_in[1];   // [N,1]
  const int*   src  = (const int*)d_in[2];     // [E]
  const int*   dst  = (const int*)d_in[3];     // [E]
  const int N = in_sizes[1];
  const int E = in_sizes[2];
  float* out = (float*)d_out;

  // Workspace carve-out (≈58.9 MB): ping buffer + CSR machinery.
  char* ws = (char*)d_ws;
  size_t off = 0;
  auto carve = [&](size_t bytes) -> char* {
    char* p = ws + off;
    off = (off + bytes + 255) & ~(size_t)255;
    return p;
  };
  float* bufA   = (float*)carve((size_t)N * D_FEAT * sizeof(float));
  int*   rowptr = (int*)carve((size_t)(N + 1) * sizeof(int));
  int*   cursor = (int*)carve((size_t)N * sizeof(int));
  int*   deg    = (int*)carve((size_t)N * sizeof(int));
  int*   csr    = (int*)carve((size_t)E * sizeof(int));

  // 1) Build deterministic CSR-by-destination (fresh every call).
  zero_i32<<<(N + 255) / 256, 256, 0, stream>>>(deg, N);
  build_deg<<<(E + 255) / 256, 256, 0, stream>>>(dst, deg, E);
  scan_deg<<<1, 1024, 0, stream>>>(deg, rowptr, cursor, N);
  fill_csr<<<(E + 255) / 256, 256, 0, stream>>>(dst, cursor, csr, E);
  sort_resolve<<<(N + 255) / 256, 256, 0, stream>>>(rowptr, csr, src, N);

  // 2) K=10 fused, atomic-free propagation iterations (ping-pong bufA <-> d_out,
  //    final iteration writes d_out).
  const float alpha = 0.1f;
  const float* cur = h_in;
  for (int t = 0; t < 10; ++t) {
    float* nxt = (t % 2 == 0) ? bufA : out;
    prop<<<(N + WPB - 1) / WPB, WPB * 32, 0, stream>>>(cur, h_in, norm, rowptr,
                                                       csr, nxt, N, alpha);
    cur = nxt;
  }
}